// QNetwork_63582695850605
// MI455X (gfx1250) — compile-verified
//
#include <hip/hip_runtime.h>
#include <hip/hip_bf16.h>
#include <math.h>

// ---------------------------------------------------------------------------
// 12-qubit state-vector simulator, one workgroup (8 wave32) per batch element.
// State (4096 complex = 32KB) lives in LDS for the entire circuit.
// Per layer: fuse RZ*RY*RX per qubit, Kronecker 4 qubits -> 16x16 complex U,
// apply as (16x16) x (16x256) f32 matmuls via V_WMMA_F32_16X16X4_F32.
// Ring-of-CNOTs layer collapses to one fixed GF(2) index permutation.
// ---------------------------------------------------------------------------

#define NQ   12
#define NL   5
#define DIMQ 4096

typedef float v2f __attribute__((ext_vector_type(2)));
typedef float v8f __attribute__((ext_vector_type(8)));

__global__ __launch_bounds__(256)
void qsim_kernel(const float* __restrict__ x,      // [8192,12]
                 const float* __restrict__ iw,     // [5,12]
                 const float* __restrict__ th,     // [5,12,2]
                 const float* __restrict__ ow,     // [2]
                 float* __restrict__ out)          // [8192,2]
{
    __shared__ float s_re[DIMQ];
    __shared__ float s_im[DIMQ];
    __shared__ float u_re[256];   // current group's 16x16 unitary (row-major)
    __shared__ float u_im[256];
    __shared__ float u_ni[256];   // -Im(U)  (f32 WMMA has no A-negate)
    __shared__ float m_re[NQ * 4];// per-qubit fused 2x2: [q][i][j] at q*4+i*2+j
    __shared__ float m_im[NQ * 4];
    __shared__ float red[2];

    const int b    = blockIdx.x;
    const int tid  = threadIdx.x;
    const int lane = tid & 31;
    const int wave = tid >> 5;      // 0..7
    const int ln   = lane & 15;     // lane's M (A) / N (B,D) index
    const int hb   = lane >> 4;     // half-of-wave selector

    // ---- init state |00..0> ----
    for (int i = tid; i < DIMQ; i += 256) { s_re[i] = 0.0f; s_im[i] = 0.0f; }
    if (tid == 0) s_re[0] = 1.0f;
    __syncthreads();

    for (int l = 0; l < NL; ++l) {
        // ---- build per-qubit fused gate M = RZ(b)*RY(a)*RX(t) ----
        if (tid < NQ) {
            const int j = tid;
            const float t  = 0.5f * iw[l * NQ + j] * tanhf(x[b * NQ + j]);
            const float c  = cosf(t),  s  = sinf(t);
            const float a  = 0.5f * th[(l * NQ + j) * 2 + 0];
            const float ca = cosf(a),  sa = sinf(a);
            const float bb = 0.5f * th[(l * NQ + j) * 2 + 1];
            const float cb = cosf(bb), sb = sinf(bb);
            // T = RY*RX
            const float t00r =  ca * c, t00i =  sa * s;
            const float t01r = -sa * c, t01i = -ca * s;
            const float t10r =  sa * c, t10i = -ca * s;
            const float t11r =  ca * c, t11i = -sa * s;
            // M = diag(cb - i sb, cb + i sb) * T
            m_re[j*4+0] = cb*t00r + sb*t00i;  m_im[j*4+0] = cb*t00i - sb*t00r;
            m_re[j*4+1] = cb*t01r + sb*t01i;  m_im[j*4+1] = cb*t01i - sb*t01r;
            m_re[j*4+2] = cb*t10r - sb*t10i;  m_im[j*4+2] = cb*t10i + sb*t10r;
            m_re[j*4+3] = cb*t11r - sb*t11i;  m_im[j*4+3] = cb*t11i + sb*t11r;
        }
        __syncthreads();

        // ---- three 4-qubit groups: U_g = M(4g) (x) M(4g+1) (x) M(4g+2) (x) M(4g+3)
        for (int g = 0; g < 3; ++g) {
            // build one U entry per thread (256 entries)
            {
                const int r = tid >> 4, cc = tid & 15;
                float ur = 1.0f, ui = 0.0f;
                #pragma unroll
                for (int k = 0; k < 4; ++k) {
                    const int q  = 4 * g + k;
                    const int bi = (r  >> (3 - k)) & 1;
                    const int bj = (cc >> (3 - k)) & 1;
                    const float mr = m_re[q*4 + bi*2 + bj];
                    const float mi = m_im[q*4 + bi*2 + bj];
                    const float nr = ur * mr - ui * mi;
                    ui = ur * mi + ui * mr;
                    ur = nr;
                }
                u_re[tid] = ur; u_im[tid] = ui; u_ni[tid] = -ui;
            }
            __syncthreads();

            const int stride = (g == 0) ? 256 : (g == 1) ? 16 : 1;

            // A fragments (16x4 f32, per ISA lane layout), uniform per wave/group
            v2f aR[4], aI[4], aN[4];
            #pragma unroll
            for (int kc = 0; kc < 4; ++kc) {
                const int ko = 4 * kc + 2 * hb;
                aR[kc].x = u_re[ln * 16 + ko];  aR[kc].y = u_re[ln * 16 + ko + 1];
                aI[kc].x = u_im[ln * 16 + ko];  aI[kc].y = u_im[ln * 16 + ko + 1];
                aN[kc].x = u_ni[ln * 16 + ko];  aN[kc].y = u_ni[ln * 16 + ko + 1];
            }

            // 16 column-tiles of 16 columns; waves own disjoint tiles -> in-place
            for (int t = wave; t < 16; t += 8) {
                const int col = t * 16 + ln;
                int base;
                if (g == 0)      base = col;
                else if (g == 1) base = ((col >> 4) << 8) | (col & 15);
                else             base = col << 4;

                v2f bR[4], bI[4];
                #pragma unroll
                for (int kc = 0; kc < 4; ++kc) {
                    const int ko = 4 * kc + 2 * hb;
                    bR[kc].x = s_re[base +  ko      * stride];
                    bR[kc].y = s_re[base + (ko + 1) * stride];
                    bI[kc].x = s_im[base +  ko      * stride];
                    bI[kc].y = s_im[base + (ko + 1) * stride];
                }

                v8f dre = {0.f,0.f,0.f,0.f,0.f,0.f,0.f,0.f};
                v8f dim = {0.f,0.f,0.f,0.f,0.f,0.f,0.f,0.f};
                #pragma unroll
                for (int kc = 0; kc < 4; ++kc) {
                    // D_re += Ur*Sre + (-Ui)*Sim ; D_im += Ui*Sre + Ur*Sim
                    dre = __builtin_amdgcn_wmma_f32_16x16x4_f32(
                              false, aR[kc], false, bR[kc], (short)0, dre, false, false);
                    dre = __builtin_amdgcn_wmma_f32_16x16x4_f32(
                              false, aN[kc], false, bI[kc], (short)0, dre, false, false);
                    dim = __builtin_amdgcn_wmma_f32_16x16x4_f32(
                              false, aI[kc], false, bR[kc], (short)0, dim, false, false);
                    dim = __builtin_amdgcn_wmma_f32_16x16x4_f32(
                              false, aR[kc], false, bI[kc], (short)0, dim, false, false);
                }

                #pragma unroll
                for (int v = 0; v < 8; ++v) {
                    const int m = v + 8 * hb;
                    s_re[base + m * stride] = dre[v];
                    s_im[base + m * stride] = dim[v];
                }
            }
            __syncthreads();
        }

        // ---- CNOT ring as one fixed permutation (register-staged scatter) ----
        float pr[16], pi[16];
        #pragma unroll
        for (int k = 0; k < 16; ++k) {
            const int i = tid + 256 * k;
            pr[k] = s_re[i]; pi[k] = s_im[i];
        }
        __syncthreads();
        #pragma unroll
        for (int k = 0; k < 16; ++k) {
            const int i = tid + 256 * k;
            int y = i;
            y ^= y >> 1; y ^= y >> 2; y ^= y >> 4; y ^= y >> 8;
            const int j = (y & 0x7FF) | (((y ^ (i >> 11)) & 1) << 11);
            s_re[j] = pr[k]; s_im[j] = pi[k];
        }
        __syncthreads();
    }

    // ---- expectation values <Z0 Z1>, <Z2 Z3> ----
    if (tid < 2) red[tid] = 0.0f;
    __syncthreads();
    float e01 = 0.0f, e23 = 0.0f;
    for (int i = tid; i < DIMQ; i += 256) {
        const float p = s_re[i] * s_re[i] + s_im[i] * s_im[i];
        e01 += (((i >> 11) ^ (i >> 10)) & 1) ? -p : p;   // qubits 0,1 = bits 11,10
        e23 += (((i >> 9)  ^ (i >> 8))  & 1) ? -p : p;   // qubits 2,3 = bits 9,8
    }
    atomicAdd(&red[0], e01);
    atomicAdd(&red[1], e23);
    __syncthreads();
    if (tid == 0) {
        out[b * 2 + 0] = ow[0] * red[0];
        out[b * 2 + 1] = ow[1] * red[1];
    }
}

extern "C" void kernel_launch(void* const* d_in, const int* in_sizes, int n_in,
                              void* d_out, int out_size, void* d_ws, size_t ws_size,
                              hipStream_t stream) {
    (void)n_in; (void)d_ws; (void)ws_size; (void)out_size;
    const float* x  = (const float*)d_in[0];   // [B,12]
    const float* iw = (const float*)d_in[1];   // [5,12]
    const float* th = (const float*)d_in[2];   // [5,12,2]
    const float* ow = (const float*)d_in[3];   // [2]
    float* out = (float*)d_out;
    const int batch = in_sizes[0] / NQ;        // 8192
    qsim_kernel<<<batch, 256, 0, stream>>>(x, iw, th, ow, out);
}